// Capture_Data_62972810494257
// MI455X (gfx1250) — compile-verified
//
#include <hip/hip_runtime.h>
#include <hip/hip_bf16.h>
#include <stdint.h>

typedef float v2f __attribute__((ext_vector_type(2)));
typedef float v8f __attribute__((ext_vector_type(8)));

#define KDIM   112          // c * s = 4 * 28
#define PLANE  65536        // h * w = 256 * 256
#define CONS_EPS 1e-10f
#define NL_IN  0.5f
#define POISSON_T (20.0f * NL_IN)   // 10.0

// ---------------------------------------------------------------------------
// Kernel 1: cap = sum_k spec*filt, via WMMA f32 16x16x4 with B == ones.
// One wave handles 16 consecutive pixels. A-operand layout (ISA 7.12.2):
//   lanes 0-15  : M = lane,    VGPR0 -> K=0, VGPR1 -> K=1
//   lanes 16-31 : M = lane-16, VGPR0 -> K=2, VGPR1 -> K=3
// So lane L (<16) streams planes (k, k+1) at pixel L, lane L+16 streams
// planes (k+2, k+3) at the same pixel. C/D layout: lane 0 VGPR r = cap[pix r],
// lane 16 VGPR r = cap[pix 8+r].  Writes peak = cap + eps into d_out[P .. 2P).
// ---------------------------------------------------------------------------
__global__ __launch_bounds__(256)
void capture_cap_wmma_kernel(const float* __restrict__ spec,
                             const float* __restrict__ filt,
                             float* __restrict__ out,
                             int n_groups, int n_pixels)
{
    const int tid   = blockIdx.x * blockDim.x + threadIdx.x;
    const int wave  = tid >> 5;            // one 16-pixel group per wave
    const int lane  = tid & 31;
    if (wave >= n_groups) return;          // uniform per wave

    const int pg  = wave << 4;             // first global pixel of the group
    const int b   = pg / PLANE;            // batch (group never crosses batch: PLANE % 16 == 0)
    const int pix = pg - b * PLANE;        // pixel offset inside image

    const int row   = lane & 15;           // pixel within group
    const int khalf = (lane >> 4) << 1;    // 0 for lanes 0-15, 2 for lanes 16-31

    // flat index of (b, k=khalf, pix+row)
    size_t off = ((size_t)b * KDIM + (size_t)khalf) * PLANE + (size_t)pix + (size_t)row;

    v8f c = {};                            // f32 16x16 accumulator (8 VGPRs)
    v2f bones = {1.0f, 1.0f};              // B = ones -> WMMA acts as K-reduction

#pragma unroll
    for (int i = 0; i < KDIM / 4; ++i) {
        float s0 = spec[off];
        float s1 = spec[off + (size_t)PLANE];
        float f0 = filt[off];
        float f1 = filt[off + (size_t)PLANE];
        v2f a = { s0 * f0, s1 * f1 };
        // D = A(16x4) * ones(4x16) + C : each row of D = partial sum for that pixel
        c = __builtin_amdgcn_wmma_f32_16x16x4_f32(
                /*neg_a=*/false, a, /*neg_b=*/false, bones,
                /*c_mod=*/(short)0, c, /*reuse_a=*/false, /*reuse_b=*/false);
        off += (size_t)4 * PLANE;
    }

    // Lane 0 holds cap for pixels pg+0..7 in c[0..7]; lane 16 for pg+8..15.
    if ((lane & 15) == 0) {
        size_t base = (size_t)n_pixels + (size_t)pg + (size_t)((lane >> 4) << 3);
#pragma unroll
        for (int r = 0; r < 8; ++r)
            out[base + r] = c[r] + CONS_EPS;   // peak segment
    }
}

// ---------------------------------------------------------------------------
// Deterministic counter-based RNG (PCG32) + Poisson / Gaussian samplers.
// ---------------------------------------------------------------------------
__device__ __forceinline__ uint32_t pcg_next(uint64_t& s)
{
    s = s * 6364136223846793005ULL + 1442695040888963407ULL;
    uint32_t x   = (uint32_t)(((s >> 18) ^ s) >> 27);
    uint32_t rot = (uint32_t)(s >> 59);
    return (x >> rot) | (x << ((32u - rot) & 31u));
}

__device__ __forceinline__ float pcg_uf(uint64_t& s)   // uniform in (0, 1]
{
    return ((pcg_next(s) >> 8) + 1u) * (1.0f / 16777216.0f);
}

__device__ __forceinline__ float poisson_knuth(uint64_t& s, float lambda)
{
    float L = __expf(-lambda);
    float p = 1.0f;
    int   k = 0;
    while (p > L && k < 64) { p *= pcg_uf(s); ++k; }
    return (float)(k > 0 ? k - 1 : 0);
}

__device__ __forceinline__ float gauss_bm(uint64_t& s)
{
    float u1 = pcg_uf(s);
    float u2 = pcg_uf(s);
    return sqrtf(-2.0f * __logf(u1)) * __cosf(6.28318530718f * u2);
}

__device__ __forceinline__ float poisson_any(uint64_t& s, float lambda)
{
    if (lambda < 12.0f) return poisson_knuth(s, lambda);
    float r = rintf(lambda + sqrtf(lambda) * gauss_bm(s));  // normal approx
    return r < 0.0f ? 0.0f : r;
}

// ---------------------------------------------------------------------------
// Kernel 2: per-pixel noise synthesis. Reads peak (segment 1, written by
// kernel 1 earlier on the same stream), writes noisy / dark_t / gauss_t^2.
// ---------------------------------------------------------------------------
__global__ __launch_bounds__(256)
void capture_noise_kernel(float* __restrict__ out, int n_pixels)
{
    int idx = blockIdx.x * blockDim.x + threadIdx.x;
    if (idx >= n_pixels) return;

    float peak = out[(size_t)n_pixels + (size_t)idx];

    uint64_t s = (uint64_t)idx * 0x9E3779B97F4A7C15ULL + 0xD1B54A32D192ED03ULL;
    (void)pcg_next(s);                         // warm the stream

    float pn = poisson_any(s, peak);           // Poisson(cap + eps)
    float dn = poisson_knuth(s, 1.0f * NL_IN); // Poisson(0.5)
    float gn = gauss_bm(s) * (1.0f * NL_IN);   // N(0,1) * 0.5

    float noisy = (pn + dn + gn) * (POISSON_T / 255.0f);

    out[idx] = noisy;                                        // segment 0: noisy
    out[(size_t)2 * n_pixels + idx] = NL_IN;                 // segment 2: dark_t
    out[(size_t)3 * n_pixels + idx] = NL_IN * NL_IN;         // segment 3: gauss_t^2
}

// ---------------------------------------------------------------------------
extern "C" void kernel_launch(void* const* d_in, const int* in_sizes, int n_in,
                              void* d_out, int out_size, void* d_ws, size_t ws_size,
                              hipStream_t stream)
{
    const float* spec = (const float*)d_in[0];
    const float* filt = (const float*)d_in[1];
    float* out = (float*)d_out;

    const int n_pixels = in_sizes[0] / KDIM;       // 524288 = 8*256*256
    const int n_groups = n_pixels / 16;            // 16 pixels per wave

    // Kernel 1: one wave32 per group -> n_groups*32 threads
    {
        const int threads = 256;
        const int blocks  = (n_groups * 32 + threads - 1) / threads;
        capture_cap_wmma_kernel<<<blocks, threads, 0, stream>>>(
            spec, filt, out, n_groups, n_pixels);
    }

    // Kernel 2: one thread per pixel
    {
        const int threads = 256;
        const int blocks  = (n_pixels + threads - 1) / threads;
        capture_noise_kernel<<<blocks, threads, 0, stream>>>(out, n_pixels);
    }
}